// DALayer_55113020342863
// MI455X (gfx1250) — compile-verified
//
#include <hip/hip_runtime.h>
#include <hip/hip_bf16.h>

// ---------------------------------------------------------------------------
// SE-with-expert-routing layer for MI455X (gfx1250, wave32).
//
// Perf model: memory-bound. 512MiB read (pool) + 512MiB read + 512MiB write
// (scale) = 1.5GiB @ 23.3TB/s ~= 69us floor. GEMMs are ~12.6 MFLOP total and
// are done with V_WMMA_F32_16X16X4_F32 in a single small workgroup.
// ---------------------------------------------------------------------------

typedef __attribute__((ext_vector_type(2))) float v2f;
typedef __attribute__((ext_vector_type(4))) float v4f;
typedef __attribute__((ext_vector_type(8))) float v8f;

#define BB    64
#define CC    512
#define HIDN  32
#define NEXP  3
#define PLANE 4096   // H*W = 64*64

// ---------------------------------------------------------------------------
// Kernel 1: global average pool.  One wave per (b,c) plane (4096 floats).
// Each lane does 32 x b128 non-temporal loads (512B/step per wave, coalesced),
// then a 5-step shfl_xor reduction (wave32).
// ---------------------------------------------------------------------------
__global__ __launch_bounds__(256) void se_pool(const float* __restrict__ x,
                                               float* __restrict__ y)
{
    const int wave = threadIdx.x >> 5;
    const int lane = threadIdx.x & 31;
    const int p    = blockIdx.x * 8 + wave;          // plane id in [0, B*C)

    const v4f* px = (const v4f*)(x + (size_t)p * PLANE);
    float s = 0.f;
#pragma unroll
    for (int i = 0; i < 32; ++i) {
        v4f v = __builtin_nontemporal_load(px + i * 32 + lane);
        s += v.x + v.y + v.z + v.w;
    }
#pragma unroll
    for (int off = 16; off > 0; off >>= 1)
        s += __shfl_xor(s, off, 32);
    if (lane == 0)
        y[p] = s * (1.0f / (float)PLANE);
}

// ---------------------------------------------------------------------------
// Kernel 2: gate = sigmoid( relu(y @ W1[d]^T) @ W2[d]^T ), per-sample expert d.
// Single workgroup, 16 waves. Both GEMMs via V_WMMA_F32_16X16X4_F32.
// Expert routing: compute every expert's candidate tile (3x redundancy on a
// ~MFLOP problem is free) and select rows by dataset[b] in the epilogue, so
// the GEMM itself stays dense and EXEC stays all-ones around WMMA.
//
// fp32 WMMA fragment layout (ISA 7.12.2):
//   A (16x4): lane L holds row M=L%16; v0=K0+2*(L/16), v1=K0+1+2*(L/16)
//   B (4x16): mirrored (col N=L%16, same K split)
//   C/D (16x16): vgpr j: M = j + 8*(L/16), N = L%16
// ---------------------------------------------------------------------------
__global__ __launch_bounds__(512) void se_gate(const float* __restrict__ y,
                                               const int*   __restrict__ dataset,
                                               const float* __restrict__ W1,
                                               const float* __restrict__ W2,
                                               float*       __restrict__ gate)
{
    __shared__ float h_all[NEXP][BB][HIDN];   // 24 KB
    __shared__ float h_sel[BB][HIDN];         //  8 KB
    __shared__ int   ds[BB];

    const int tid   = threadIdx.x;
    const int wave  = tid >> 5;        // 0..15
    const int lane  = tid & 31;
    const int lm    = lane & 15;       // row/col within 16
    const int khalf = lane >> 4;       // 0 or 1: which K pair this lane holds

    if (tid < BB) ds[tid] = dataset[tid];

    // ---- Stage 1: h_all[e] = relu(y @ W1[e]^T).  M=64, N=32, K=512.
    // Tiles: e(3) x mt(4) x nt(2) = 24, round-robin over 16 waves.
    for (int tile = wave; tile < NEXP * 8; tile += 16) {
        const int e  = tile >> 3;
        const int r  = tile & 7;
        const int mt = r >> 1;
        const int nt = r & 1;
        const int m  = mt * 16 + lm;
        const int n  = nt * 16 + lm;
        const float* ya = y  + (size_t)m * CC;
        const float* wb = W1 + ((size_t)e * HIDN + n) * CC;

        v8f acc = {};
#pragma unroll 4
        for (int k = 0; k < CC; k += 4) {
            const int kk = k + khalf * 2;
            v2f a = *(const v2f*)(ya + kk);
            v2f b = *(const v2f*)(wb + kk);
            acc = __builtin_amdgcn_wmma_f32_16x16x4_f32(
                false, a, false, b, (short)0, acc, false, false);
        }
#pragma unroll
        for (int j = 0; j < 8; ++j) {
            const int row = mt * 16 + j + khalf * 8;
            const float v = acc[j];
            h_all[e][row][nt * 16 + lm] = v > 0.f ? v : 0.f;
        }
    }
    __syncthreads();

    // ---- Stage 2: per-row expert selection into h_sel.
    for (int i = tid; i < BB * HIDN; i += 512) {
        const int b = i >> 5, n = i & 31;
        h_sel[b][n] = h_all[ds[b]][b][n];
    }
    __syncthreads();

    // ---- Stage 3: gate = sigmoid(h_sel @ W2[e]^T), rows selected by expert.
    // M=64, N=512, K=32.  Tiles: e(3) x mt(4) x nt(32) = 384.
    for (int tile = wave; tile < NEXP * 128; tile += 16) {
        const int e  = tile >> 7;
        const int r  = tile & 127;
        const int mt = r >> 5;
        const int nt = r & 31;
        const int m  = mt * 16 + lm;
        const int c  = nt * 16 + lm;
        const float* wb = W2 + ((size_t)e * CC + c) * HIDN;

        v8f acc = {};
#pragma unroll
        for (int k = 0; k < HIDN; k += 4) {
            const int kk = k + khalf * 2;
            v2f a = *(const v2f*)(&h_sel[m][kk]);   // ds_load_b64
            v2f b = *(const v2f*)(wb + kk);
            acc = __builtin_amdgcn_wmma_f32_16x16x4_f32(
                false, a, false, b, (short)0, acc, false, false);
        }
#pragma unroll
        for (int j = 0; j < 8; ++j) {
            const int row = mt * 16 + j + khalf * 8;
            if (ds[row] == e) {
                const float v = acc[j];
                gate[(size_t)row * CC + nt * 16 + lm] = 1.0f / (1.0f + __expf(-v));
            }
        }
    }
}

// ---------------------------------------------------------------------------
// Kernel 3: out = x * gate[b,c].  Each block covers 1024 contiguous floats
// (quarter of a plane), so the gate index is block-uniform -> scalar load.
// b128 non-temporal load/store for the 1 GiB stream.
// ---------------------------------------------------------------------------
__global__ __launch_bounds__(256) void se_scale(const float* __restrict__ x,
                                                const float* __restrict__ gate,
                                                float*       __restrict__ out)
{
    const float g = gate[blockIdx.x >> 2];           // plane id, uniform per block
    const size_t t = (size_t)blockIdx.x * 256 + threadIdx.x;
    v4f v = __builtin_nontemporal_load((const v4f*)x + t);
    v *= g;
    __builtin_nontemporal_store(v, (v4f*)out + t);
}

// ---------------------------------------------------------------------------
extern "C" void kernel_launch(void* const* d_in, const int* in_sizes, int n_in,
                              void* d_out, int out_size, void* d_ws, size_t ws_size,
                              hipStream_t stream)
{
    const float* x       = (const float*)d_in[0];   // [64,512,64,64]
    const int*   dataset = (const int*)  d_in[1];   // [64]
    const float* W1      = (const float*)d_in[2];   // [3,32,512]
    const float* W2      = (const float*)d_in[3];   // [3,512,32]
    float*       out     = (float*)d_out;           // [64,512,64,64]

    float* y    = (float*)d_ws;                     // [64*512] pooled means
    float* gate = y + BB * CC;                      // [64*512] gates  (256 KB total ws)

    // 1) pool: 32768 planes, 8 planes (waves) per 256-thread block
    se_pool<<<(BB * CC) / 8, 256, 0, stream>>>(x, y);

    // 2) tiny routed double-GEMM + activations, all WMMA, one workgroup
    se_gate<<<1, 512, 0, stream>>>(y, dataset, W1, W2, gate);

    // 3) scale: 2^25 float4s, one per thread
    se_scale<<<(size_t)(BB * CC * PLANE) / (256 * 4), 256, 0, stream>>>(x, gate, out);
}